// GCNNetwork_51977694216539
// MI455X (gfx1250) — compile-verified
//
#include <hip/hip_runtime.h>
#include <hip/hip_bf16.h>
#include <stdint.h>

// ---------------------------------------------------------------------------
// GCN inference on MI455X (gfx1250, wave32).
//   layer0: hw0 = x @ W0            (100000x512 @ 512x16)  -- fp32 WMMA 16x16x4
//   agg0   = segsum(norm * hw0[src] -> dst) (+self loop)   -- f32 atomics in L2
//   h      = relu(agg0 + b0)
//   hw1    = h @ W1                 (100000x16 @ 16x32)    -- fp32 WMMA 16x16x4
//   out    = segsum(norm * hw1[src] -> dst) + b1 (+self loop)
// ---------------------------------------------------------------------------

typedef float v2f __attribute__((ext_vector_type(2)));
typedef float v8f __attribute__((ext_vector_type(8)));

__device__ __forceinline__ void atomAddF32(float* p, float v) {
  unsafeAtomicAdd(p, v);   // -> global_atomic_add_f32 (L2 atomic unit)
}

// ---------------- degree / normalization -----------------------------------

__global__ void k_deg_init(float* deg, int n) {
  int i = blockIdx.x * blockDim.x + threadIdx.x;
  if (i < n) deg[i] = 1.0f;                       // self-loop contribution
}

__global__ void k_deg_count(const long long* __restrict__ ei, float* deg, int e) {
  int t = blockIdx.x * blockDim.x + threadIdx.x;
  if (t < e) atomAddF32(&deg[(int)ei[(size_t)e + t]], 1.0f);  // dst = ei[1][t]
}

__global__ void k_dinv(float* deg, int n) {
  int i = blockIdx.x * blockDim.x + threadIdx.x;
  if (i < n) deg[i] = __frsqrt_rn(fmaxf(deg[i], 1.0f));       // in-place -> dinv
}

// ---------------- layer 0 GEMM: hw0 = x @ W0 (K=512, out 16) ---------------
// One wave per 16-row tile. A: 16x4 f32 frag, lane=(M=lane&15, Kpair=2*(lane>>4)).

__global__ void k_gemm0_wmma(const float* __restrict__ x,
                             const float* __restrict__ W0,
                             float* __restrict__ hw0, int nTiles) {
  const int wave = threadIdx.x >> 5;
  const int lane = threadIdx.x & 31;
  const int tile = blockIdx.x * (blockDim.x >> 5) + wave;
  if (tile >= nTiles) return;                     // wave-uniform: EXEC stays all-1s

  const int m    = lane & 15;                     // A row / D col index
  const int half = lane >> 4;                     // K-pair selector
  const float* __restrict__ arow = x  + (size_t)(tile * 16 + m) * 512 + 2 * half;
  const float* __restrict__ bcol = W0 + m + 2 * half * 16;   // W0[kk*16 + n]

  v8f acc = {};
  for (int k = 0; k < 512; k += 16) {
    __builtin_prefetch(arow + k + 16, 0, 0);      // global_prefetch_b8, next 64B of row
#pragma unroll
    for (int kk = 0; kk < 16; kk += 4) {
      v2f a, b;
      a.x = arow[k + kk];
      a.y = arow[k + kk + 1];
      b.x = bcol[(k + kk) * 16];
      b.y = bcol[(k + kk) * 16 + 16];
      acc = __builtin_amdgcn_wmma_f32_16x16x4_f32(
          false, a, false, b, (short)0, acc, false, false);
    }
  }
  // D layout: lane n = lane&15, VGPR v -> row M = v + 8*half
  float* orow = hw0 + (size_t)(tile * 16) * 16 + m;
#pragma unroll
  for (int v = 0; v < 8; ++v)
    orow[(v + 8 * half) * 16] = acc[v];
}

// ---------------- layer 0 aggregation ---------------------------------------

// agg0 init = self-loop term: hw0[i]*dinv[i]^2   (also zero-initializes)
__global__ void k_self16(const float* __restrict__ hw0,
                         const float* __restrict__ dinv,
                         float* __restrict__ agg0, int n) {
  int g = blockIdx.x * blockDim.x + threadIdx.x;
  if (g < n * 16) {
    float di = dinv[g >> 4];
    agg0[g] = hw0[g] * di * di;
  }
}

// one thread = (edge, float4 feature chunk); 4 chunks for H=16
__global__ void k_edge16(const long long* __restrict__ ei,
                         const float* __restrict__ dinv,
                         const float* __restrict__ hw0,
                         float* __restrict__ agg0, int e) {
  int g = blockIdx.x * blockDim.x + threadIdx.x;
  int eid = g >> 2;
  if (eid >= e) return;
  int c = (g & 3) * 4;
  int s = (int)ei[eid];
  int d = (int)ei[(size_t)e + eid];
  float nrm = dinv[s] * dinv[d];
  float4 v = *(const float4*)(hw0 + (size_t)s * 16 + c);   // gather: hits L2
  float* o = agg0 + (size_t)d * 16 + c;
  atomAddF32(o + 0, v.x * nrm);
  atomAddF32(o + 1, v.y * nrm);
  atomAddF32(o + 2, v.z * nrm);
  atomAddF32(o + 3, v.w * nrm);
}

// ---------------- layer 1 GEMM: hw1 = relu(agg0+b0) @ W1 (K=16, out 32) -----

__global__ void k_gemm1_wmma(const float* __restrict__ agg0,
                             const float* __restrict__ b0,
                             const float* __restrict__ W1,
                             float* __restrict__ hw1, int nTiles) {
  const int wave = threadIdx.x >> 5;
  const int lane = threadIdx.x & 31;
  const int tile = blockIdx.x * (blockDim.x >> 5) + wave;
  if (tile >= nTiles) return;

  const int m    = lane & 15;
  const int half = lane >> 4;
  const float* __restrict__ arow = agg0 + (size_t)(tile * 16 + m) * 16 + 2 * half;

  v8f acc0 = {};   // output cols 0..15
  v8f acc1 = {};   // output cols 16..31
#pragma unroll
  for (int k = 0; k < 16; k += 4) {
    int kk = k + 2 * half;
    v2f a, bA, bB;
    a.x = fmaxf(arow[k]     + b0[kk],     0.0f);   // fused bias + ReLU
    a.y = fmaxf(arow[k + 1] + b0[kk + 1], 0.0f);
    bA.x = W1[kk * 32 + m];
    bA.y = W1[(kk + 1) * 32 + m];
    bB.x = W1[kk * 32 + 16 + m];
    bB.y = W1[(kk + 1) * 32 + 16 + m];
    acc0 = __builtin_amdgcn_wmma_f32_16x16x4_f32(
        false, a, false, bA, (short)0, acc0, false, false);
    acc1 = __builtin_amdgcn_wmma_f32_16x16x4_f32(
        false, a, false, bB, (short)0, acc1, false, false);
  }
  float* orow = hw1 + (size_t)(tile * 16) * 32 + m;
#pragma unroll
  for (int v = 0; v < 8; ++v) {
    orow[(v + 8 * half) * 32]      = acc0[v];
    orow[(v + 8 * half) * 32 + 16] = acc1[v];
  }
}

// ---------------- layer 1 aggregation ---------------------------------------

// out init = b1 + self-loop term
__global__ void k_outinit32(const float* __restrict__ hw1,
                            const float* __restrict__ dinv,
                            const float* __restrict__ b1,
                            float* __restrict__ out, int n) {
  int g = blockIdx.x * blockDim.x + threadIdx.x;
  if (g < n * 32) {
    float di = dinv[g >> 5];
    out[g] = b1[g & 31] + hw1[g] * di * di;
  }
}

// one thread = (edge, float4 chunk); 8 chunks for C=32
__global__ void k_edge32(const long long* __restrict__ ei,
                         const float* __restrict__ dinv,
                         const float* __restrict__ hw1,
                         float* __restrict__ out, int e) {
  int g = blockIdx.x * blockDim.x + threadIdx.x;
  int eid = g >> 3;
  if (eid >= e) return;
  int c = (g & 7) * 4;
  int s = (int)ei[eid];
  int d = (int)ei[(size_t)e + eid];
  float nrm = dinv[s] * dinv[d];
  float4 v = *(const float4*)(hw1 + (size_t)s * 32 + c);
  float* o = out + (size_t)d * 32 + c;
  atomAddF32(o + 0, v.x * nrm);
  atomAddF32(o + 1, v.y * nrm);
  atomAddF32(o + 2, v.z * nrm);
  atomAddF32(o + 3, v.w * nrm);
}

// ---------------------------------------------------------------------------

extern "C" void kernel_launch(void* const* d_in, const int* in_sizes, int n_in,
                              void* d_out, int out_size, void* d_ws, size_t ws_size,
                              hipStream_t stream) {
  const float*     x  = (const float*)d_in[0];
  const long long* ei = (const long long*)d_in[1];   // int64 [2, E]
  const float*     W0 = (const float*)d_in[2];       // [512, 16]
  const float*     b0 = (const float*)d_in[3];       // [16]
  const float*     W1 = (const float*)d_in[4];       // [16, 32]
  const float*     b1 = (const float*)d_in[5];       // [32]
  float*           out = (float*)d_out;              // [N, 32]

  const int N = in_sizes[0] / 512;
  const int E = in_sizes[1] / 2;

  // workspace carve-up (all regions fit in L2: ~26 MB total)
  float* ws = (float*)d_ws;
  auto au = [](size_t v) { return (v + 127) & ~(size_t)127; };
  size_t off = 0;
  float* dinv = ws + off; off += au((size_t)N);        // deg -> dinv in place
  float* hw0  = ws + off; off += au((size_t)N * 16);
  float* agg0 = ws + off; off += au((size_t)N * 16);
  float* hw1  = ws + off;

  const int B = 256;
  const int nTiles = (N + 15) / 16;          // N = 100000 -> 6250 exact
  const int wavesPerBlk = B / 32;

  // normalization
  k_deg_init <<<(N + B - 1) / B, B, 0, stream>>>(dinv, N);
  k_deg_count<<<(E + B - 1) / B, B, 0, stream>>>(ei, dinv, E);
  k_dinv     <<<(N + B - 1) / B, B, 0, stream>>>(dinv, N);

  // layer 0
  k_gemm0_wmma<<<(nTiles + wavesPerBlk - 1) / wavesPerBlk, B, 0, stream>>>(x, W0, hw0, nTiles);
  k_self16    <<<((size_t)N * 16 + B - 1) / B, B, 0, stream>>>(hw0, dinv, agg0, N);
  k_edge16    <<<((size_t)E * 4 + B - 1) / B, B, 0, stream>>>(ei, dinv, hw0, agg0, E);

  // layer 1
  k_gemm1_wmma<<<(nTiles + wavesPerBlk - 1) / wavesPerBlk, B, 0, stream>>>(agg0, b0, W1, hw1, nTiles);
  k_outinit32 <<<((size_t)N * 32 + B - 1) / B, B, 0, stream>>>(hw1, dinv, b1, out, N);
  k_edge32    <<<((size_t)E * 8 + B - 1) / B, B, 0, stream>>>(ei, dinv, hw1, out, E);
}